// sample_interpolate_57140244906534
// MI455X (gfx1250) — compile-verified
//
#include <hip/hip_runtime.h>

typedef float v2f __attribute__((ext_vector_type(2)));
typedef float v4f __attribute__((ext_vector_type(4)));
typedef float v8f __attribute__((ext_vector_type(8)));

#define IMG_H 256
#define IMG_W 256
#define IMG_C 64
#define PIX_PER_BATCH (IMG_H * IMG_W)

// Each wave32: 1 WMMA computes x_s,y_s for 16 consecutive pixels (the reference's
// einsum theta[2,3] x [gx;gy;1] done as 16x16x4 f32 WMMA), then 8 sampling
// iterations of 2 pixels each (16 lanes x float4 = one pixel's 64 channels).
__global__ __launch_bounds__(256) void stn_bilinear_kernel(
    const float* __restrict__ X,      // [8,256,256,64]
    const float* __restrict__ theta,  // [8,6]
    float* __restrict__ out)          // [8,256,256,64]
{
    const int b     = blockIdx.y;
    const int lane  = threadIdx.x & 31;
    const int wave  = threadIdx.x >> 5;
    const int wavePix = ((blockIdx.x << 3) + wave) << 4;   // first of 16 pixels
    const int py  = wavePix >> 8;          // row (uniform across the wave's 16 px)
    const int px0 = wavePix & 255;         // first column (16-aligned)

    // theta for this batch (b uniform per block -> scalar loads)
    const float t0 = theta[b * 6 + 0], t1 = theta[b * 6 + 1], t2 = theta[b * 6 + 2];
    const float t3 = theta[b * 6 + 3], t4 = theta[b * 6 + 4], t5 = theta[b * 6 + 5];

    // ---- Grid generation via V_WMMA_F32_16X16X4_F32 ----
    // A (16x4):  row0 = [t0 t1 t2 0] (x_s), row1 = [t3 t4 t5 0] (y_s), rest 0.
    //   f32 A layout: lanes 0-15 hold K=0 (v0) / K=1 (v1) for M=lane;
    //                 lanes 16-31 hold K=2 (v0) / K=3 (v1) for M=lane-16.
    // B (4x16):  col n = [gx_n, gy, 1, 0].
    //   f32 B layout (symmetric): lanes 0-15 hold K=0 (v0) / K=1 (v1) for N=lane;
    //                 lanes 16-31 hold K=2 (v0) / K=3 (v1) for N=lane-16.
    const float gy  = -1.0f + (2.0f / 255.0f) * (float)py;
    const int   n   = lane & 15;
    const float gxn = -1.0f + (2.0f / 255.0f) * (float)(px0 + n);

    v2f amat, bmat;
    if (lane < 16) {
        amat.x = (n == 0) ? t0 : (n == 1) ? t3 : 0.0f;   // K=0
        amat.y = (n == 0) ? t1 : (n == 1) ? t4 : 0.0f;   // K=1
        bmat.x = gxn;                                     // K=0, col n
        bmat.y = gy;                                      // K=1, col n
    } else {
        amat.x = (n == 0) ? t2 : (n == 1) ? t5 : 0.0f;   // K=2
        amat.y = 0.0f;                                    // K=3
        bmat.x = 1.0f;                                    // K=2, col n
        bmat.y = 0.0f;                                    // K=3, col n
    }
    v8f dacc = {};
    // D row M=0 -> x_s (dacc[0], lanes 0..15), row M=1 -> y_s (dacc[1], lanes 0..15)
    dacc = __builtin_amdgcn_wmma_f32_16x16x4_f32(false, amat, false, bmat,
                                                 (short)0, dacc, false, false);

    // ---- Bilinear sampling: 2 pixels per wave iteration ----
    const float* __restrict__ Xb = X + (size_t)b * (PIX_PER_BATCH * IMG_C);
    float* __restrict__ Ob = out + (size_t)b * (PIX_PER_BATCH * IMG_C)
                                 + (size_t)wavePix * IMG_C;
    const int c    = (lane & 15) << 2;   // channel offset (float index)
    const int half = lane >> 4;          // 0: pixel 2i, 1: pixel 2i+1

    for (int i = 0; i < 8; ++i) {
        const int p = (i << 1) + half;
        const float xs = __shfl(dacc[0], p, 32);
        const float ys = __shfl(dacc[1], p, 32);

        // Reference: x = 0.5*((xs+1)*(max_x-1)), max_x = 255
        const float x = 0.5f * (xs + 1.0f) * 254.0f;
        const float y = 0.5f * (ys + 1.0f) * 254.0f;
        const int ix0 = (int)floorf(x);
        const int iy0 = (int)floorf(y);
        const int x0 = min(max(ix0, 0), IMG_W - 1);
        const int x1 = min(max(ix0 + 1, 0), IMG_W - 1);
        const int y0 = min(max(iy0, 0), IMG_H - 1);
        const int y1 = min(max(iy0 + 1, 0), IMG_H - 1);

        const float dx1 = (float)x1 - x, dx0 = x - (float)x0;
        const float dy1 = (float)y1 - y, dy0 = y - (float)y0;
        const float wa = dx1 * dy1;
        const float wb = dx1 * dy0;
        const float wc = dx0 * dy1;
        const float wd = dx0 * dy0;

        // 4 coalesced 128-bit corner loads (temporal: input is L2-resident)
        const v4f Ia = *(const v4f*)(Xb + ((size_t)(y0 * IMG_W + x0) * IMG_C) + c);
        const v4f Ib = *(const v4f*)(Xb + ((size_t)(y1 * IMG_W + x0) * IMG_C) + c);
        const v4f Ic = *(const v4f*)(Xb + ((size_t)(y0 * IMG_W + x1) * IMG_C) + c);
        const v4f Id = *(const v4f*)(Xb + ((size_t)(y1 * IMG_W + x1) * IMG_C) + c);

        v4f r;
        r.x = fmaf(wd, Id.x, fmaf(wc, Ic.x, fmaf(wb, Ib.x, wa * Ia.x)));
        r.y = fmaf(wd, Id.y, fmaf(wc, Ic.y, fmaf(wb, Ib.y, wa * Ia.y)));
        r.z = fmaf(wd, Id.z, fmaf(wc, Ic.z, fmaf(wb, Ib.z, wa * Ia.z)));
        r.w = fmaf(wd, Id.w, fmaf(wc, Ic.w, fmaf(wb, Ib.w, wa * Ia.w)));

        // Non-temporal 128-bit store: output streams once, keep input in L2
        __builtin_nontemporal_store(r, (v4f*)(Ob + (size_t)p * IMG_C + c));
    }
}

extern "C" void kernel_launch(void* const* d_in, const int* in_sizes, int n_in,
                              void* d_out, int out_size, void* d_ws, size_t ws_size,
                              hipStream_t stream) {
    const float* X     = (const float*)d_in[0];   // [8,256,256,64] f32
    const float* theta = (const float*)d_in[1];   // [8,6] f32
    float* out = (float*)d_out;

    // 8 waves/block, 16 px/wave -> 128 px/block; 65536 px/batch -> 512 blocks; y = batch
    dim3 grid(PIX_PER_BATCH / 128, 8, 1);
    dim3 block(256, 1, 1);
    stn_bilinear_kernel<<<grid, block, 0, stream>>>(X, theta, out);
}